// GatedGraphConvWoGRU_51625506898539
// MI455X (gfx1250) — compile-verified
//
#include <hip/hip_runtime.h>

// ---------------- problem constants (from reference) ----------------
#define IN_FEATS   96
#define OUT_FEATS  128
#define N_ETYPES   4
// K dimension of the factored GEMM: 4 etypes * 96 feats + 4 count rows
#define K_DIM      (N_ETYPES * IN_FEATS + N_ETYPES)   // 388, divisible by 4
#define K_PAIRS    (K_DIM / 2)                        // 194

typedef __attribute__((ext_vector_type(2))) float v2f;
typedef __attribute__((ext_vector_type(4))) float v4f;
typedef __attribute__((ext_vector_type(8))) float v8f;

// ---------------- kernel 1: zero the accumulator table ----------------
__global__ __launch_bounds__(256) void ggc_zero(float* __restrict__ p, size_t n) {
    size_t i = (size_t)blockIdx.x * blockDim.x + threadIdx.x;
    size_t stride = (size_t)gridDim.x * blockDim.x;
    size_t n4 = n >> 2;
    v4f* p4 = (v4f*)p;
    v4f z = {0.f, 0.f, 0.f, 0.f};
    for (size_t k = i; k < n4; k += stride) p4[k] = z;
    for (size_t k = (n4 << 2) + i; k < n; k += stride) p[k] = 0.f;
}

// ---------------- kernel 2: pack pair-interleaved B ----------------
// Logical B[kk][j], kk in [0,388): kk<384 -> t=kk/96,k=kk%96 : W[t][j][k]; kk>=384 -> b[kk-384][j]
// Stored as Bm2[p][j][2] with kk = 2p + h  =>  fragment (kk, kk+1) is one b64 per lane.
__global__ __launch_bounds__(256) void ggc_packB(const float* __restrict__ W,
                                                 const float* __restrict__ b,
                                                 float* __restrict__ Bm2) {
    int idx = blockIdx.x * blockDim.x + threadIdx.x;   // (p*128 + j)*2 + h
    if (idx >= K_DIM * OUT_FEATS) return;
    int h  = idx & 1;
    int j  = (idx >> 1) & (OUT_FEATS - 1);
    int p  = idx >> 8;
    int kk = 2 * p + h;
    float v;
    if (kk < N_ETYPES * IN_FEATS) {
        int t = kk / IN_FEATS;
        int k = kk - t * IN_FEATS;
        v = W[((size_t)t * OUT_FEATS + j) * OUT_FEATS + k];
    } else {
        v = b[(kk - N_ETYPES * IN_FEATS) * OUT_FEATS + j];
    }
    Bm2[idx] = v;
}

// ---------------- kernel 3: edge scatter into pair-interleaved sums ----------------
// Logical row kk = t*96+k : per-node sum of feature k over type-t in-edges;
// row 384+t : type-t in-degree (float). Stored as sT2[p][node][2], kk = 2p + h.
// One wave32 per edge: lane l covers feature columns l, l+32, l+64 (coalesced gather).
__global__ __launch_bounds__(256) void ggc_scatter(const float* __restrict__ feat,
                                                   const int*   __restrict__ etypes,
                                                   const int*   __restrict__ src,
                                                   const int*   __restrict__ dst,
                                                   float* __restrict__ sT2,
                                                   int n_edges, int n_nodes) {
    int e    = (int)((blockIdx.x * (unsigned)blockDim.x + threadIdx.x) >> 5);
    int lane = threadIdx.x & 31;
    if (e >= n_edges) return;
    int t = etypes[e];
    int s = src[e];
    int d = dst[e];
    const float* frow = feat + (size_t)s * IN_FEATS;
    size_t nn = (size_t)n_nodes;
    #pragma unroll
    for (int c = 0; c < IN_FEATS; c += 32) {
        float v = frow[c + lane];
        int row = t * IN_FEATS + c + lane;
        atomicAdd(&sT2[((size_t)(row >> 1) * nn + d) * 2 + (row & 1)], v);
    }
    if (lane == 0) {
        int row = N_ETYPES * IN_FEATS + t;
        atomicAdd(&sT2[((size_t)(row >> 1) * nn + d) * 2 + (row & 1)], 1.0f);
    }
}

// ---------------- kernel 4: WMMA f32 GEMM  out = A(50000x388) @ B(388x128) ----------------
// A fragment (M=lane&15, K pair) and B fragment (N=lane&15, K pair) are each a single
// contiguous b64 load thanks to the interleaved layouts. 97 K-steps, 8 WMMAs per step.
__global__ __launch_bounds__(256) void ggc_gemm(const float* __restrict__ sT2,
                                                const float* __restrict__ Bm2,
                                                float* __restrict__ out,
                                                int n_nodes, int n_tiles) {
    int lane  = threadIdx.x & 31;
    int wave  = threadIdx.x >> 5;
    int tile  = blockIdx.x * 8 + wave;
    if (tile >= n_tiles) return;

    int node0 = tile * 16;
    int m     = lane & 15;          // A: M index, B: N index, C/D: N index
    int hi    = lane >> 4;          // 0: K pair kk0/2, 1: K pair kk0/2 + 1
    int mrow  = node0 + m;
    if (mrow > n_nodes - 1) mrow = n_nodes - 1;   // clamp partial tile (stores guarded)
    size_t nn = (size_t)n_nodes;

    v8f acc[8] = {};

    for (int p0 = 0; p0 < K_PAIRS; p0 += 2) {
        int p = p0 + hi;            // pair holds rows (2p, 2p+1): lanes 0-15 -> kk0,kk0+1; 16-31 -> kk0+2,kk0+3
        v2f a = *(const v2f*)(sT2 + ((size_t)p * nn + mrow) * 2);
        const float* Bp = Bm2 + (size_t)p * (OUT_FEATS * 2);
        #pragma unroll
        for (int nt = 0; nt < 8; nt++) {
            v2f bv = *(const v2f*)(Bp + (nt * 16 + m) * 2);
            acc[nt] = __builtin_amdgcn_wmma_f32_16x16x4_f32(
                /*neg_a=*/false, a, /*neg_b=*/false, bv,
                /*c_mod=*/(short)0, acc[nt], /*reuse_a=*/false, /*reuse_b=*/false);
        }
    }

    // C/D layout: VGPR r -> M = r + 8*(lane>=16), N = lane&15
    int mbase = node0 + hi * 8;
    #pragma unroll
    for (int r = 0; r < 8; r++) {
        int node = mbase + r;
        if (node < n_nodes) {
            float* orow = out + (size_t)node * OUT_FEATS;
            #pragma unroll
            for (int nt = 0; nt < 8; nt++) {
                orow[nt * 16 + m] = acc[nt][r];
            }
        }
    }
}

// ---------------- launcher ----------------
extern "C" void kernel_launch(void* const* d_in, const int* in_sizes, int n_in,
                              void* d_out, int out_size, void* d_ws, size_t ws_size,
                              hipStream_t stream) {
    const float* feat   = (const float*)d_in[0];
    const int*   etypes = (const int*)  d_in[1];
    const int*   src    = (const int*)  d_in[2];
    const int*   dst    = (const int*)  d_in[3];
    const float* W      = (const float*)d_in[4];
    const float* b      = (const float*)d_in[5];
    float*       out    = (float*)d_out;

    int n_nodes = in_sizes[0] / IN_FEATS;   // 50000
    int n_edges = in_sizes[1];              // 500000

    // workspace: sT2 [K_PAIRS][n_nodes][2] floats, then Bm2 [K_PAIRS][128][2] (~77.8 MB)
    float* sT2 = (float*)d_ws;
    float* Bm2 = sT2 + (size_t)K_DIM * n_nodes;

    size_t s_elems = (size_t)K_DIM * n_nodes;
    ggc_zero<<<2048, 256, 0, stream>>>(sT2, s_elems);

    int packN = K_DIM * OUT_FEATS;
    ggc_packB<<<(packN + 255) / 256, 256, 0, stream>>>(W, b, Bm2);

    int scatter_blocks = (n_edges * 32 + 255) / 256;   // one wave32 per edge
    ggc_scatter<<<scatter_blocks, 256, 0, stream>>>(feat, etypes, src, dst, sT2,
                                                    n_edges, n_nodes);

    int n_tiles = (n_nodes + 15) / 16;                 // 3125
    ggc_gemm<<<(n_tiles + 7) / 8, 256, 0, stream>>>(sT2, Bm2, out, n_nodes, n_tiles);
}